// DPLossV2_25683904430206
// MI455X (gfx1250) — compile-verified
//
#include <hip/hip_runtime.h>
#include <hip/hip_bf16.h>

// ---------------------------------------------------------------------------
// DPLossV2 for gfx1250 (MI455X, wave32, WMMA)
//
// Stage 1: row squared-norms of x and z.
// Stage 2: dist_x / dist_z via V_WMMA_F32_16X16X4_F32 Gram tiles (exact f32).
// Stage 3: soft-rank loss via native V_TANH_F32:
//            sigmoid(-d) = 0.5*(1 - tanh(d/2)); constants cancel in the
//            rank difference, leaving diff = 0.5*(Sum tanh_x - Sum tanh_z).
//          One block per row i, rows pre-halved + interleaved in LDS.
// Stage 4: reduce partials, write (total, rank_loss, 0).
// ---------------------------------------------------------------------------

typedef __attribute__((ext_vector_type(2))) float v2f;
typedef __attribute__((ext_vector_type(8))) float v8f;

#define N_PTS 512

// Native gfx1250 tanh (TRANS32 pipe). Builtin if available, else inline asm
// with v_nops to cover the TRANS co-execution RAW hazard, else exp fallback.
static __device__ __forceinline__ float tanh_hw(float x) {
#if __has_builtin(__builtin_amdgcn_tanhf)
    return __builtin_amdgcn_tanhf(x);
#elif defined(__gfx1250__)
    float r;
    asm volatile("v_tanh_f32 %0, %1\n\tv_nop\n\tv_nop" : "=v"(r) : "v"(x));
    return r;
#else
    float e = __expf(2.0f * x);
    return (e - 1.0f) / (e + 1.0f);
#endif
}

// ---- Stage 1: per-row squared norms --------------------------------------
__global__ void rowsq_kernel(const float* __restrict__ X, float* __restrict__ sq,
                             int cols) {
    int r = blockIdx.x * blockDim.x + threadIdx.x;
    if (r < N_PTS) {
        const float* p = X + (size_t)r * cols;
        float s = 0.0f;
        #pragma unroll 4
        for (int c = 0; c < cols; ++c) {
            float v = p[c];
            s += v * v;
        }
        sq[r] = s;
    }
}

// ---- Stage 2: Gram tile via WMMA f32 16x16x4, then euclidean dist --------
// One wave (32 threads) per 16x16 output tile. 32x32 tiles -> 1024 blocks.
// A layout (16x4, MxK): lane L holds row M=L&15; VGPR v holds K = 2*(L>>4)+v.
// B layout (4x16, KxN): lane L holds col N=L&15; VGPR v holds K = 2*(L>>4)+v.
// For a Gram matrix both operands load with the SAME pattern (different base
// row), and the two K elements per lane are contiguous -> single b64 load.
__global__ void dist_wmma_kernel(const float* __restrict__ X,
                                 const float* __restrict__ sq,
                                 float* __restrict__ dist,
                                 int K /* row length: 512 or 64 */) {
    const int lane   = threadIdx.x;        // 0..31, one wave
    const int l15    = lane & 15;
    const int half   = lane >> 4;          // 0 or 1
    const int tileId = blockIdx.x;
    const int tr     = tileId >> 5;        // tile row   (0..31)
    const int tc     = tileId & 31;        // tile col   (0..31)

    const int rowBase = tr * 16;
    const int colBase = tc * 16;

    const float* Arow = X + (size_t)(rowBase + l15) * K + 2 * half;
    const float* Brow = X + (size_t)(colBase + l15) * K + 2 * half;

    v8f c = {};
    #pragma unroll 4
    for (int k = 0; k < K; k += 4) {
        v2f a = *(const v2f*)(Arow + k);
        v2f b = *(const v2f*)(Brow + k);
        // D = A * B + C   (16x16x4, pure f32 — exact vs reference)
        c = __builtin_amdgcn_wmma_f32_16x16x4_f32(
                /*neg_a=*/false, a, /*neg_b=*/false, b,
                /*c_mod=*/(short)0, c, /*reuse_a=*/false, /*reuse_b=*/false);
    }

    // C/D layout: VGPR r -> M = r + 8*half, N = l15.
    const float sqn = sq[colBase + l15];
    #pragma unroll
    for (int r = 0; r < 8; ++r) {
        int m = r + 8 * half;
        float g  = c[r];
        float d2 = sq[rowBase + m] + sqn - 2.0f * g;
        d2 = fmaxf(d2, 0.0f);
        float d = (d2 > 0.0f) ? __builtin_sqrtf(d2) : 0.0f;
        dist[(size_t)(rowBase + m) * N_PTS + (colBase + l15)] = d;
    }
}

// ---- Stage 3: soft-rank squared-difference, one block per row i ----------
// rank_z[i,j]-rank_x[i,j] = 0.5 * sum_k [ tanh((mx_ij-mx_ik)/2)
//                                       - tanh((mz_ij-mz_ik)/2) ]
// Rows are stored PRE-HALVED and interleaved (x,z) in LDS, so the inner loop
// is one broadcast ds_load_b64 + 4x (v_sub, v_tanh, v_add) per k.
__global__ __launch_bounds__(256)
void rankloss_kernel(const float* __restrict__ dx,
                     const float* __restrict__ dz,
                     float* __restrict__ partials) {
    __shared__ float2 sm[N_PTS];   // sm[k] = {0.5*dist_x[i,k], 0.5*dist_z[i,k]}
    __shared__ float  red[256];

    const int i   = blockIdx.x;
    const int tid = threadIdx.x;

    for (int t = tid; t < N_PTS; t += 256) {
        float2 v;
        v.x = 0.5f * dx[(size_t)i * N_PTS + t];
        v.y = 0.5f * dz[(size_t)i * N_PTS + t];
        sm[t] = v;
    }
    __syncthreads();

    const int j0 = tid;          // columns owned by this thread
    const int j1 = tid + 256;
    const float hx0 = sm[j0].x, hz0 = sm[j0].y;
    const float hx1 = sm[j1].x, hz1 = sm[j1].y;

    float sx0 = 0.0f, sz0 = 0.0f, sx1 = 0.0f, sz1 = 0.0f;
    #pragma unroll 4
    for (int k = 0; k < N_PTS; ++k) {
        const float2 v = sm[k];            // LDS broadcast (all lanes same addr)
        sx0 += tanh_hw(hx0 - v.x);
        sz0 += tanh_hw(hz0 - v.y);
        sx1 += tanh_hw(hx1 - v.x);
        sz1 += tanh_hw(hz1 - v.y);
    }
    const float d0 = 0.5f * (sx0 - sz0);   // = rank_z - rank_x (up to sign)
    const float d1 = 0.5f * (sx1 - sz1);
    float acc = d0 * d0 + d1 * d1;

    red[tid] = acc;
    __syncthreads();
    #pragma unroll
    for (int s = 128; s > 0; s >>= 1) {
        if (tid < s) red[tid] += red[tid + s];
        __syncthreads();
    }
    if (tid == 0) partials[i] = red[0];
}

// ---- Stage 4: final reduction + output -----------------------------------
__global__ __launch_bounds__(256)
void finalize_kernel(const float* __restrict__ partials, float* __restrict__ out) {
    __shared__ float red[256];
    const int tid = threadIdx.x;
    float s = partials[tid] + partials[tid + 256];
    red[tid] = s;
    __syncthreads();
    #pragma unroll
    for (int st = 128; st > 0; st >>= 1) {
        if (tid < st) red[tid] += red[tid + st];
        __syncthreads();
    }
    if (tid == 0) {
        const float rank_loss = red[0] * (1.0f / ((float)N_PTS * (float)N_PTS));
        out[0] = rank_loss;   // total  (LAMBDA_RANK=1, LAMBDA_PAIRDIST=0)
        out[1] = rank_loss;   // rank_loss
        out[2] = 0.0f;        // pairdist_loss
    }
}

// ---------------------------------------------------------------------------
extern "C" void kernel_launch(void* const* d_in, const int* in_sizes, int n_in,
                              void* d_out, int out_size, void* d_ws, size_t ws_size,
                              hipStream_t stream) {
    const float* x = (const float*)d_in[0];   // [512, 512]
    const float* z = (const float*)d_in[1];   // [512, 64]
    float* out = (float*)d_out;               // 3 floats

    // Workspace layout (floats)
    float* ws       = (float*)d_ws;
    float* dist_x   = ws;                        // 512*512
    float* dist_z   = dist_x + N_PTS * N_PTS;    // 512*512
    float* sqx      = dist_z + N_PTS * N_PTS;    // 512
    float* sqz      = sqx + N_PTS;               // 512
    float* partials = sqz + N_PTS;               // 512

    // Stage 1: squared norms
    rowsq_kernel<<<2, 256, 0, stream>>>(x, sqx, 512);
    rowsq_kernel<<<2, 256, 0, stream>>>(z, sqz, 64);

    // Stage 2: pairwise distances (WMMA Gram tiles), 32x32 tiles of 16x16
    dist_wmma_kernel<<<1024, 32, 0, stream>>>(x, sqx, dist_x, 512);
    dist_wmma_kernel<<<1024, 32, 0, stream>>>(z, sqz, dist_z, 64);

    // Stage 3: soft-rank loss partials (one block per row)
    rankloss_kernel<<<N_PTS, 256, 0, stream>>>(dist_x, dist_z, partials);

    // Stage 4: reduce + write outputs
    finalize_kernel<<<1, 256, 0, stream>>>(partials, out);
}